// QAttentionLayer_65481071407082
// MI455X (gfx1250) — compile-verified
//
#include <hip/hip_runtime.h>

// out = cumprod(cos(x + params), axis=-1), x: [32, 8192, 64] f32, params: [64] f32
// Memory-bound streaming scan: 128 MB traffic @ 23.3 TB/s ~ 5.5 us floor.
// One half-wave (16 lanes) per 64-float row; float4 per lane; multiplicative
// Hillis-Steele scan across the 16-lane segment via width-16 shuffles.
// Data movement: double-buffered gfx1250 GLOBAL_LOAD_ASYNC_TO_LDS_B128 pipeline
// (one async DMA instruction per wave per tile, tracked with ASYNCcnt).

typedef float f4  __attribute__((ext_vector_type(4)));
typedef int   v4i __attribute__((ext_vector_type(4)));

#define QUBITS 64
#define WAVES_PER_BLOCK 8   // 256 threads / wave32

#if defined(__gfx1250__) && __has_builtin(__builtin_amdgcn_global_load_async_to_lds_b128)
#define HAVE_ASYNC_LDS 1
typedef __attribute__((address_space(1))) v4i g_v4i;
typedef __attribute__((address_space(3))) v4i l_v4i;

// One 16B async copy per lane -> one global_load_async_to_lds_b128 per wave.
// AS1 pointers share the generic 64-bit VA; an AS3 pointer is the low 32 bits
// of a generic LDS pointer (aperture lives in the high bits).
__device__ __forceinline__ void async_copy_b128(const float* gsrc, float* ldst) {
  __builtin_amdgcn_global_load_async_to_lds_b128(
      (g_v4i*)(unsigned long long)gsrc,
      (l_v4i*)(unsigned int)(unsigned long long)ldst,
      /*offset=*/0, /*cpol=RT (working set fits 192MB L2)*/ 0);
}

#if __has_builtin(__builtin_amdgcn_s_wait_asynccnt)
#define WAIT_ASYNC(N)                           \
  do {                                          \
    __builtin_amdgcn_s_wait_asynccnt(N);        \
    __asm__ volatile("" ::: "memory");          \
  } while (0)
#else
#define WAIT_ASYNC(N)                                         \
  __asm__ volatile("s_wait_asynccnt %0" ::"i"(N) : "memory")
#endif
#endif  // HAVE_ASYNC_LDS

__global__ __launch_bounds__(256)
void qcumprod_kernel(const float* __restrict__ x,
                     const float* __restrict__ params,
                     float* __restrict__ out,
                     int nPairs) {
  const int lane   = threadIdx.x & 31;
  const int sub    = lane & 15;   // float4 slot within the row (cols 4*sub..4*sub+3)
  const int rowSel = lane >> 4;   // which of the wave's two rows this half-wave handles

  const int wave       = (blockIdx.x * blockDim.x + threadIdx.x) >> 5;
  const int waveStride = (gridDim.x * blockDim.x) >> 5;

  // params broadcast: both half-waves read the same 4 floats for their column slot
  const f4 p = *(const f4*)(params + sub * 4);

#if defined(HAVE_ASYNC_LDS)
  // Double-buffered staging: [buf][wave-in-block][2 rows x 64 floats] = 8 KB/block
  __shared__ float stage[2][WAVES_PER_BLOCK][2 * QUBITS];
  const int wIB = threadIdx.x >> 5;
  float* l0 = &stage[0][wIB][lane * 4];   // this lane's 16B slot (row-major 2x64 tile)
  float* l1 = &stage[1][wIB][lane * 4];

  int pair = wave;
  int buf  = 0;
  if (pair < nPairs) {
    const float* src = x + (((long long)pair << 1) + rowSel) * QUBITS + sub * 4;
    async_copy_b128(src, l0);             // prologue: fill buffer 0
  }

  for (; pair < nPairs; pair += waveStride) {
    const int next = pair + waveStride;
    if (next < nPairs) {
      // Issue next tile into the other buffer, then wait for the current one.
      const float* nsrc = x + (((long long)next << 1) + rowSel) * QUBITS + sub * 4;
      async_copy_b128(nsrc, buf ? l0 : l1);
      WAIT_ASYNC(1);                      // async loads retire in order per wave
    } else {
      WAIT_ASYNC(0);                      // drain for the final tile
    }

    const float* cur = buf ? l1 : l0;
    f4 v = *(const f4*)cur;               // ds_load_b128 from staged tile
#else
  for (int pair = wave; pair < nPairs; pair += waveStride) {
    const long long rowD = ((long long)pair << 1) + rowSel;
    if (pair + waveStride < nPairs)       // fallback: register path + prefetch
      __builtin_prefetch(x + (rowD + (long long)waveStride * 2) * QUBITS + sub * 4, 0, 3);
    f4 v = *(const f4*)(x + rowD * QUBITS + sub * 4);
#endif

    // Fast hardware cosine (v_cos_f32, TRANS pipe): inputs are ~N(0,2), tiny range.
    float c0 = __cosf(v[0] + p[0]);
    float c1 = __cosf(v[1] + p[1]);
    float c2 = __cosf(v[2] + p[2]);
    float c3 = __cosf(v[3] + p[3]);

    // In-lane prefix products of the 4 local factors.
    float pp0 = c0;
    float pp1 = pp0 * c1;
    float pp2 = pp1 * c2;
    float pp3 = pp2 * c3;   // lane-total

    // Multiplicative inclusive scan of lane-totals across the 16-lane segment.
    float t = pp3;
#pragma unroll
    for (int d = 1; d < 16; d <<= 1) {
      float u = __shfl_up(t, (unsigned)d, 16);
      t *= (sub >= d) ? u : 1.0f;
    }
    // Exclusive prefix for this lane = inclusive scan shifted up by one lane.
    float ex = __shfl_up(t, 1u, 16);
    if (sub == 0) ex = 1.0f;

    const long long row = ((long long)pair << 1) + rowSel;
    f4 o;
    o[0] = ex * pp0;
    o[1] = ex * pp1;
    o[2] = ex * pp2;
    o[3] = ex * pp3;
    *(f4*)(out + row * QUBITS + sub * 4) = o;

#if defined(HAVE_ASYNC_LDS)
    buf ^= 1;
#endif
  }
}

extern "C" void kernel_launch(void* const* d_in, const int* in_sizes, int n_in,
                              void* d_out, int out_size, void* d_ws, size_t ws_size,
                              hipStream_t stream) {
  const float* x      = (const float*)d_in[0];  // [32, 8192, 64] f32
  const float* params = (const float*)d_in[1];  // [64] f32
  float* out          = (float*)d_out;          // [32, 8192, 64] f32

  const long long rows = (long long)out_size / QUBITS;  // 262144
  const int nPairs = (int)(rows >> 1);                  // 2 rows per wave per iter

  const int block = 256;                                // 8 waves (wave32)
  long long gridLL = (nPairs + WAVES_PER_BLOCK - 1) / WAVES_PER_BLOCK;
  if (gridLL > 4096) gridLL = 4096;   // grid-stride: ~4 iters/wave -> pipeline has depth
  const int grid = (int)gridLL;

  qcumprod_kernel<<<grid, block, 0, stream>>>(x, params, out, nPairs);
}